// SpectralAutoencoder_31533649887385
// MI455X (gfx1250) — compile-verified
//
#include <hip/hip_runtime.h>
#include <hip/hip_bf16.h>

// -------------------------------------------------------------------------
// GCN (2 layers, H=64) + mean pool + linear head for MI455X (gfx1250).
// Heavy part = edge gather/scatter (L2-resident, f32 atomics).
// Dense GEMM x1(131072x64) @ W2(64x64) runs on V_WMMA_F32_16X16X4_F32,
// full f32 precision (this workload is L2-bound, not matrix-pipe bound).
// -------------------------------------------------------------------------

typedef float v2f __attribute__((ext_vector_type(2)));
typedef float v8f __attribute__((ext_vector_type(8)));

#define HID 64
#define NB  64   // batches

// ---- degree / dinv ------------------------------------------------------
__global__ void gcn_init_deg(float* deg, int N) {
  int i = blockIdx.x * blockDim.x + threadIdx.x;
  if (i < N) deg[i] = 1.0f;                      // deg = segsum(1) + 1
}

__global__ void gcn_scatter_deg(const int* __restrict__ col, float* deg, int E) {
  int e = blockIdx.x * blockDim.x + threadIdx.x;
  if (e < E) atomicAdd(&deg[col[e]], 1.0f);
}

__global__ void gcn_finalize_dinv(float* dinv, int N) {
  int i = blockIdx.x * blockDim.x + threadIdx.x;
  if (i < N) dinv[i] = rsqrtf(dinv[i]);
}

// ---- layer-1 projection: h = x[:,None] @ W1  (outer product) ------------
__global__ void gcn_h1(const float* __restrict__ x, const float* __restrict__ W1,
                       float* __restrict__ h, int N) {
  int idx = blockIdx.x * blockDim.x + threadIdx.x;
  if (idx < N * HID) {
    int i = idx >> 6, j = idx & 63;
    h[idx] = x[i] * W1[j];
  }
}

// ---- agg init: agg = dinv^2 * h + bias  (self term + bias pre-folded) ---
__global__ void gcn_init_agg(const float* __restrict__ dinv, const float* __restrict__ h,
                             const float* __restrict__ bias, float* __restrict__ agg, int N) {
  int idx = blockIdx.x * blockDim.x + threadIdx.x;
  if (idx < N * HID) {
    int i = idx >> 6, j = idx & 63;
    float d = dinv[i];
    agg[idx] = d * d * h[idx] + bias[j];
  }
}

// ---- edge scatter: agg[col] += dinv[row]*dinv[col] * h[row] -------------
__global__ void gcn_edge_scatter(const int* __restrict__ row, const int* __restrict__ col,
                                 const float* __restrict__ dinv, const float* __restrict__ h,
                                 float* __restrict__ agg, long long E) {
  long long total  = E << 6;                     // E * 64 features
  long long stride = (long long)gridDim.x * blockDim.x;
  for (long long idx = (long long)blockIdx.x * blockDim.x + threadIdx.x;
       idx < total; idx += stride) {
    long long e = idx >> 6;
    int j = (int)(idx & 63);
    int r = row[e], c = col[e];
    float norm = dinv[r] * dinv[c];
    atomicAdd(&agg[(long long)c * HID + j], norm * h[(long long)r * HID + j]);
  }
}

// ---- in-place ReLU ------------------------------------------------------
__global__ void gcn_relu(float* __restrict__ a, int N) {
  int idx = blockIdx.x * blockDim.x + threadIdx.x;
  if (idx < N * HID) a[idx] = fmaxf(a[idx], 0.0f);
}

// ---- WMMA GEMM: Hout(N x 64) = X(N x 64) @ W(64 x 64), f32 --------------
// 128 threads = 4 waves; each wave owns a 16-column tile, block owns 16 rows.
// A 16x4: lane<16 -> K=0,1 / lane>=16 -> K=2,3 of row M=lane&15.
// B 4x16: mirrored (lane picks N, upper half-lanes take K=2,3).
// C/D 8 VGPRs: M = r + 8*(lane>>4), N = lane&15.
__global__ __launch_bounds__(128)
void gcn_gemm_wmma(const float* __restrict__ X, const float* __restrict__ W,
                   float* __restrict__ Hout) {
  const int lane   = threadIdx.x & 31;
  const int wave   = threadIdx.x >> 5;           // N-tile 0..3
  const int rowB   = blockIdx.x * 16;            // 16 rows per block
  const int m      = lane & 15;
  const int khalf  = (lane >> 4) << 1;           // 0 or 2
  const int ncol   = (wave << 4) + (lane & 15);

  v8f acc = {};
  #pragma unroll
  for (int kb = 0; kb < HID; kb += 4) {
    int ka = kb + khalf;
    v2f a = *(const v2f*)(X + (size_t)(rowB + m) * HID + ka);   // K, K+1 contiguous
    v2f b;
    b.x = W[(size_t)ka * HID + ncol];
    b.y = W[(size_t)(ka + 1) * HID + ncol];
    acc = __builtin_amdgcn_wmma_f32_16x16x4_f32(
        /*neg_a=*/false, a, /*neg_b=*/false, b,
        /*c_mod=*/(short)0, acc, /*reuse_a=*/false, /*reuse_b=*/false);
  }

  const int rb = rowB + ((lane >> 4) << 3);      // +8 rows for upper half-lanes
  #pragma unroll
  for (int r = 0; r < 8; ++r)
    Hout[(size_t)(rb + r) * HID + ncol] = acc[r];
}

// ---- pooling ------------------------------------------------------------
__global__ void gcn_zero_pool(float* pooled_cnt, int n) {   // B*128 sums + B counts
  int i = blockIdx.x * blockDim.x + threadIdx.x;
  if (i < n) pooled_cnt[i] = 0.0f;
}

__global__ void gcn_pool(const int* __restrict__ batch, const float* __restrict__ x1,
                         const float* __restrict__ x2, float* __restrict__ pooled,
                         long long N) {
  long long total  = N << 7;                     // N * 128
  long long stride = (long long)gridDim.x * blockDim.x;
  for (long long idx = (long long)blockIdx.x * blockDim.x + threadIdx.x;
       idx < total; idx += stride) {
    long long node = idx >> 7;
    int k = (int)(idx & 127);
    float v = (k < HID) ? x1[node * HID + k] : x2[node * HID + (k - HID)];
    atomicAdd(&pooled[(long long)batch[node] * 128 + k], v);
  }
}

__global__ void gcn_count(const int* __restrict__ batch, float* __restrict__ cnt, int N) {
  int i = blockIdx.x * blockDim.x + threadIdx.x;
  if (i < N) atomicAdd(&cnt[batch[i]], 1.0f);
}

// ---- classifier: out(64x10) = (pooled/cnt) @ Wl + bl --------------------
__global__ void gcn_classifier(const float* __restrict__ pooled, const float* __restrict__ cnt,
                               const float* __restrict__ Wl, const float* __restrict__ bl,
                               float* __restrict__ out) {
  int t = blockIdx.x * blockDim.x + threadIdx.x;
  if (t >= NB * 10) return;
  int b = t / 10, o = t % 10;
  float inv = 1.0f / fmaxf(cnt[b], 1.0f);
  float s = bl[o];
  #pragma unroll 4
  for (int k = 0; k < 2 * HID; ++k)
    s += pooled[b * 128 + k] * inv * Wl[k * 10 + o];
  out[t] = s;
}

// -------------------------------------------------------------------------
extern "C" void kernel_launch(void* const* d_in, const int* in_sizes, int n_in,
                              void* d_out, int out_size, void* d_ws, size_t ws_size,
                              hipStream_t stream) {
  const float* x     = (const float*)d_in[0];
  const int*   ei    = (const int*)  d_in[1];   // edge_index (2,E), int32 per contract
  const int*   batch = (const int*)  d_in[2];
  const float* W1    = (const float*)d_in[3];
  const float* b1    = (const float*)d_in[4];
  const float* W2    = (const float*)d_in[5];
  const float* b2    = (const float*)d_in[6];
  const float* Wl    = (const float*)d_in[7];
  const float* bl    = (const float*)d_in[8];
  float* out = (float*)d_out;

  const int N = in_sizes[0];
  const int E = in_sizes[1] / 2;
  const int* row = ei;
  const int* col = ei + E;

  // ---- workspace layout (bytes) ----
  char* ws = (char*)d_ws;
  float* dinv   = (float*)ws;                         // N
  float* pooled = dinv + N;                           // NB*128
  float* cnt    = pooled + NB * 128;                  // NB
  size_t small  = ((size_t)(N + NB * 128 + NB) * sizeof(float) + 255) & ~(size_t)255;
  float* h  = (float*)(ws + small);                   // N*64 (h1, then h2)
  float* x1 = h  + (size_t)N * HID;                   // N*64
  float* x2 = x1 + (size_t)N * HID;                   // N*64

  const int T = 256;
  const int gN   = (N + T - 1) / T;
  const int gE   = (E + T - 1) / T;
  const int gNH  = (N * HID + T - 1) / T;
  const int gGS  = 32768;                             // grid-stride scatter/pool blocks

  // degree -> dinv
  gcn_init_deg<<<gN, T, 0, stream>>>(dinv, N);
  gcn_scatter_deg<<<gE, T, 0, stream>>>(col, dinv, E);
  gcn_finalize_dinv<<<gN, T, 0, stream>>>(dinv, N);

  // layer 1
  gcn_h1<<<gNH, T, 0, stream>>>(x, W1, h, N);
  gcn_init_agg<<<gNH, T, 0, stream>>>(dinv, h, b1, x1, N);
  gcn_edge_scatter<<<gGS, T, 0, stream>>>(row, col, dinv, h, x1, (long long)E);
  gcn_relu<<<gNH, T, 0, stream>>>(x1, N);

  // layer 2: h = x1 @ W2 via f32 WMMA, then scatter
  gcn_gemm_wmma<<<N / 16, 128, 0, stream>>>(x1, W2, h);
  gcn_init_agg<<<gNH, T, 0, stream>>>(dinv, h, b2, x2, N);
  gcn_edge_scatter<<<gGS, T, 0, stream>>>(row, col, dinv, h, x2, (long long)E);
  gcn_relu<<<gNH, T, 0, stream>>>(x2, N);

  // pool + classify
  gcn_zero_pool<<<(NB * 129 + T - 1) / T, T, 0, stream>>>(pooled, NB * 129);
  gcn_pool<<<gGS, T, 0, stream>>>(batch, x1, x2, pooled, (long long)N);
  gcn_count<<<gN, T, 0, stream>>>(batch, cnt, N);
  gcn_classifier<<<(NB * 10 + T - 1) / T, T, 0, stream>>>(pooled, cnt, Wl, bl, out);
}